// MaskedChamferDistance_58308476010761
// MI455X (gfx1250) — compile-verified
//
#include <hip/hip_runtime.h>

namespace {
constexpr int kB  = 64;
constexpr int kNV = 4096;
constexpr int kNL = 128;
constexpr int kD  = 256;
constexpr int kTV = 128;              // video rows per block tile
constexpr int kKC = 64;               // K chunk staged in LDS
constexpr int kLStr = kKC + 4;        // 68 floats/row (TDM pad: 64 data + 4 pad dwords)
constexpr int kTilesPerB = kNV / kTV; // 32
constexpr int kChunks = kD / kKC;     // 4
} // namespace

typedef __attribute__((ext_vector_type(2))) float    v2f;
typedef __attribute__((ext_vector_type(8))) float    v8f;
typedef __attribute__((ext_vector_type(4))) unsigned v4u;
typedef __attribute__((ext_vector_type(4))) int      v4i;
typedef __attribute__((ext_vector_type(8))) int      v8i;

__device__ __forceinline__ unsigned encOrd(float f) {
  unsigned b = __float_as_uint(f);
  return (b & 0x80000000u) ? ~b : (b | 0x80000000u);
}
__device__ __forceinline__ float decOrd(unsigned u) {
  unsigned b = (u & 0x80000000u) ? (u & 0x7fffffffu) : ~u;
  return __uint_as_float(b);
}

// Issue one TDM 2D tile load: 128 rows x 64 floats from a row-major f32 tensor
// with row stride 256 floats, into LDS at ldsOff with 4-dword padding after each
// 64-dword row (=> 68-float LDS row stride, bank-conflict-free WMMA frag loads).
__device__ __forceinline__ void tdmLoad2D(const float* gsrc, unsigned ldsOff) {
  unsigned long long ga = (unsigned long long)(uintptr_t)gsrc;
  v4u g0;
  g0[0] = 1u;                                              // count=1 (valid user D#)
  g0[1] = ldsOff;                                          // lds_addr (bytes)
  g0[2] = (unsigned)ga;                                    // global_addr[31:0]
  g0[3] = ((unsigned)(ga >> 32) & 0x01ffffffu)             // global_addr[56:32]
          | 0x80000000u;                                   // type=2 ("image") in [127:126]
  v8i g1;
  g1[0] = (2 << 16)        // data_size = 4 bytes
        | (1 << 20)        // pad_enable
        | (5 << 22)        // pad_interval: 64 dwords
        | (3 << 25);       // pad_amount:   4 dwords
  g1[1] = kKC << 16;       // tensor_dim0[15:0] = 64 (atomic_barrier_addr = 0)
  g1[2] = kTV << 16;       // tensor_dim0[31:16]=0 | tensor_dim1[15:0] = 128
  g1[3] = kKC << 16;       // tensor_dim1[31:16]=0 | tile_dim0 = 64
  g1[4] = kTV;             // tile_dim1 = 128, tile_dim2 = 0
  g1[5] = kD;              // tensor_dim0_stride[31:0] = 256
  g1[6] = 0;               // stride hi, tensor_dim1_stride lo (unused, 2D)
  g1[7] = 0;
  v4i gz = (v4i){0, 0, 0, 0};
#if __clang_major__ >= 23
  v8i gz8 = (v8i){0, 0, 0, 0, 0, 0, 0, 0};
  __builtin_amdgcn_tensor_load_to_lds(g0, g1, gz, gz, gz8, 0);
#else
  __builtin_amdgcn_tensor_load_to_lds(g0, g1, gz, gz, 0);
#endif
}

__global__ void mcd_init(unsigned* __restrict__ gmaxU, unsigned* __restrict__ minv, int n) {
  int i = blockIdx.x * blockDim.x + threadIdx.x;
  if (i == 0) *gmaxU = 0u;
  if (i < n) minv[i] = 0xFFFFFFFFu;
}

__global__ __launch_bounds__(256) void mcd_pass1(
    const float* __restrict__ vf, const float* __restrict__ lf,
    const float* __restrict__ mk,
    unsigned* __restrict__ gmaxU, unsigned* __restrict__ minvV,
    unsigned* __restrict__ minvI, float* __restrict__ clipPart)
{
  __shared__ float vs[2][kTV * kLStr];   // double-buffered video K-chunks
  __shared__ float ls[2][kNL * kLStr];   // double-buffered lang K-chunks
  __shared__ float an[kTV];
  __shared__ float bn[kNL];
  __shared__ float mrow[kTV];
  __shared__ float rmw[kTV];
  __shared__ unsigned cmv[kNL];
  __shared__ unsigned cmi[kNL];
  __shared__ unsigned gmx;

  const int t     = threadIdx.x;
  const int b     = blockIdx.x / kTilesPerB;
  const int tile  = blockIdx.x % kTilesPerB;
  const int vbase = tile * kTV;

  if (t < kNL) { cmv[t] = 0xFFFFFFFFu; cmi[t] = 0xFFFFFFFFu; }
  if (t < kTV) mrow[t] = mk[(size_t)b * kNV + vbase + t];
  if (t == 0) gmx = 0u;

  const float* vgBase = vf + ((size_t)b * kNV + vbase) * kD;
  const float* lgBase = lf + (size_t)b * kNL * kD;

  const int w    = t >> 5;
  const int lane = t & 31;
  const int ln   = lane & 15;
  const int hf   = lane >> 4;   // which half of the wave
  const int mb   = w * 16;      // this wave's M base row

  // TDM pipeline prologue: wave 0 DMAs chunk 0 into buffer 0 (EXEC-ignored wave op;
  // must be issued by exactly one wave).
  if (w == 0) {
    tdmLoad2D(vgBase, (unsigned)(uintptr_t)&vs[0][0]);
    tdmLoad2D(lgBase, (unsigned)(uintptr_t)&ls[0][0]);
  }

  v8f acc[8];
#pragma unroll
  for (int j = 0; j < 8; ++j)
    acc[j] = (v8f){0.f, 0.f, 0.f, 0.f, 0.f, 0.f, 0.f, 0.f};

  for (int kc = 0; kc < kChunks; ++kc) {
    const float* vbuf = vs[kc & 1];
    const float* lbuf = ls[kc & 1];

    if (w == 0) {
      if (kc + 1 < kChunks) {
        // Prefetch next chunk into the other buffer (free: its readers finished
        // at the end-of-iteration barrier of kc-1), overlapping DMA with WMMA.
        tdmLoad2D(vgBase + (kc + 1) * kKC, (unsigned)(uintptr_t)&vs[(kc + 1) & 1][0]);
        tdmLoad2D(lgBase + (kc + 1) * kKC, (unsigned)(uintptr_t)&ls[(kc + 1) & 1][0]);
        // TDM completions are in-order per wave: <=2 outstanding => chunk kc landed.
        __builtin_amdgcn_s_wait_tensorcnt(2);
      } else {
        __builtin_amdgcn_s_wait_tensorcnt(0);
      }
    }
    __syncthreads();   // publish wave0's completed DMA to all waves

    // accumulate squared norms from the staged chunk (threads 0..127: video, 128..255: lang)
    if (t < kTV) {
      float s = (kc == 0) ? 0.f : an[t];
      const float* row = &vbuf[t * kLStr];
      for (int d = 0; d < kKC; d += 2) {
        float2 x = *(const float2*)&row[d];
        s += x.x * x.x + x.y * x.y;
      }
      an[t] = s;
    } else {
      int l = t - kTV;
      float s = (kc == 0) ? 0.f : bn[l];
      const float* row = &lbuf[l * kLStr];
      for (int d = 0; d < kKC; d += 2) {
        float2 x = *(const float2*)&row[d];
        s += x.x * x.x + x.y * x.y;
      }
      bn[l] = s;
    }

    // WMMA GEMM over this K chunk: each wave computes rows [mb, mb+16) x all 128 cols.
    // f32 16x16x4 layout: A lane holds row (lane&15), K = k + 2*(lane>>4) .. +1 (float2).
    // B lane holds col (lane&15), same K pair -> contiguous float2 from row-major lang.
    const float* aRow = &vbuf[(mb + ln) * kLStr + 2 * hf];
    for (int k = 0; k < kKC; k += 4) {
      v2f a = *(const v2f*)&aRow[k];
#pragma unroll
      for (int j = 0; j < 8; ++j) {
        v2f bfrag = *(const v2f*)&lbuf[(j * 16 + ln) * kLStr + k + 2 * hf];
        acc[j] = __builtin_amdgcn_wmma_f32_16x16x4_f32(
            false, a, false, bfrag, (short)0, acc[j], false, false);
      }
    }
    __syncthreads();   // release this buffer for the DMA two iterations ahead
  }

  // Epilogue: p = ||v||^2 - 2 v.l + ||l||^2 ; track row-mins, valid/invalid col-mins, global max.
  // C layout: acc[j][r] = ab[row = mb + r + 8*hf][col = j*16 + ln]
  float anr[8], mr[8];
#pragma unroll
  for (int r = 0; r < 8; ++r) {
    int row = mb + r + 8 * hf;
    anr[r] = an[row];
    mr[r]  = mrow[row];
  }
  float gl = -3.402823466e38f;
  float rmin[8];
#pragma unroll
  for (int r = 0; r < 8; ++r) rmin[r] = 3.402823466e38f;

#pragma unroll
  for (int j = 0; j < 8; ++j) {
    float bnv = bn[j * 16 + ln];
    float cv = 3.402823466e38f, ci = 3.402823466e38f;
#pragma unroll
    for (int r = 0; r < 8; ++r) {
      float p = anr[r] - 2.f * acc[j][r] + bnv;
      gl = fmaxf(gl, p);
      rmin[r] = fminf(rmin[r], p);
      if (mr[r] > 0.5f) cv = fminf(cv, p);
      else              ci = fminf(ci, p);
    }
    // combine the two wave-halves (rows mb+r vs mb+8+r) for this column
    cv = fminf(cv, __shfl_xor(cv, 16, 32));
    ci = fminf(ci, __shfl_xor(ci, 16, 32));
    if (hf == 0) {
      atomicMin(&cmv[j * 16 + ln], encOrd(cv));
      atomicMin(&cmi[j * 16 + ln], encOrd(ci));
    }
  }

  // row-min: reduce across the 16 lanes of each half (xor <= 8 stays within a half)
#pragma unroll
  for (int r = 0; r < 8; ++r) {
    float v = rmin[r];
    v = fminf(v, __shfl_xor(v, 1, 32));
    v = fminf(v, __shfl_xor(v, 2, 32));
    v = fminf(v, __shfl_xor(v, 4, 32));
    v = fminf(v, __shfl_xor(v, 8, 32));
    if (ln == 0) rmw[mb + r + 8 * hf] = v * mr[r];
  }

  gl = fmaxf(gl, __shfl_xor(gl, 1, 32));
  gl = fmaxf(gl, __shfl_xor(gl, 2, 32));
  gl = fmaxf(gl, __shfl_xor(gl, 4, 32));
  gl = fmaxf(gl, __shfl_xor(gl, 8, 32));
  gl = fmaxf(gl, __shfl_xor(gl, 16, 32));
  if (lane == 0) atomicMax(&gmx, encOrd(gl));

  __syncthreads();

  if (t < kNL) {
    atomicMin(&minvV[b * kNL + t], cmv[t]);
    atomicMin(&minvI[b * kNL + t], cmi[t]);
  }
  if (t == 0) atomicMax(gmaxU, gmx);

  // deterministic per-block clip partial (no float atomics)
  for (int s = kTV / 2; s > 0; s >>= 1) {
    if (t < s) rmw[t] += rmw[t + s];
    __syncthreads();
  }
  if (t == 0) clipPart[b * kTilesPerB + tile] = rmw[0];
}

__global__ __launch_bounds__(128) void mcd_pass2(
    const float* __restrict__ mk, const unsigned* __restrict__ gmaxU,
    const unsigned* __restrict__ minvV, const unsigned* __restrict__ minvI,
    const float* __restrict__ clipPart, float* __restrict__ out)
{
  __shared__ float s1[kNL];
  __shared__ float s2[kNL];
  const int b = blockIdx.x;
  const int t = threadIdx.x;

  const float gmax = decOrd(*gmaxU);
  const float mv = decOrd(minvV[b * kNL + t]);
  const float mi = decOrd(minvI[b * kNL + t]);
  // if no invalid rows exist, mi decodes to NaN/huge -> fminf returns mv (NaN-safe)
  s1[t] = fminf(mv, mi + gmax);

  float nv = 0.f;
  for (int i = t; i < kNV; i += kNL) nv += mk[(size_t)b * kNV + i];
  s2[t] = nv;
  __syncthreads();

  for (int s = kNL / 2; s > 0; s >>= 1) {
    if (t < s) { s1[t] += s1[t + s]; s2[t] += s2[t + s]; }
    __syncthreads();
  }
  if (t == 0) {
    float clip = 0.f;
    for (int j = 0; j < kTilesPerB; ++j) clip += clipPart[b * kTilesPerB + j];
    out[b] = s1[0] / (float)kNL + clip / s2[0];
  }
}

extern "C" void kernel_launch(void* const* d_in, const int* in_sizes, int n_in,
                              void* d_out, int out_size, void* d_ws, size_t ws_size,
                              hipStream_t stream) {
  (void)in_sizes; (void)n_in; (void)out_size; (void)ws_size;
  const float* vf = (const float*)d_in[0];   // [64,4096,256]
  const float* lf = (const float*)d_in[1];   // [64,128,256]
  const float* mk = (const float*)d_in[2];   // [64,4096]
  float* out = (float*)d_out;                // [64]

  unsigned* gmaxU    = (unsigned*)d_ws;             // [1]
  unsigned* minvV    = gmaxU + 1;                   // [64*128]
  unsigned* minvI    = minvV + kB * kNL;            // [64*128]
  float*    clipPart = (float*)(minvI + kB * kNL);  // [64*32]

  mcd_init<<<(2 * kB * kNL + 255) / 256, 256, 0, stream>>>(gmaxU, minvV, 2 * kB * kNL);
  mcd_pass1<<<kB * kTilesPerB, 256, 0, stream>>>(vf, lf, mk, gmaxU, minvV, minvI, clipPart);
  mcd_pass2<<<kB, kNL, 0, stream>>>(mk, gmaxU, minvV, minvI, clipPart, out);
}